// EdgeGraphConvolution_51505247813801
// MI455X (gfx1250) — compile-verified
//
#include <hip/hip_runtime.h>

// ---------------------------------------------------------------------------
// EdgeGraphConvolution for MI455X (gfx1250, wave32)
//   out[tgt] += edge_data[e] @ (x @ W)[src[e]]   ;  out += bias
//
// Phase 1 uses V_WMMA_F32_16X16X4_F32 (full-precision fp32 WMMA) for x@W.
// Phase 2 streams edge_data (512 MB, the bandwidth bottleneck) with
// non-temporal loads and scatters with global f32 atomics.
// ---------------------------------------------------------------------------

typedef __attribute__((ext_vector_type(2))) float v2f;
typedef __attribute__((ext_vector_type(4))) float vf4;
typedef __attribute__((ext_vector_type(8))) float v8f;

#define IN_F  128
#define OUT_F 16

// ---- Phase 0: out[n][i] = bias[i] ------------------------------------------
__global__ void init_bias_kernel(float* __restrict__ out,
                                 const float* __restrict__ bias,
                                 int total) {
    int idx = blockIdx.x * blockDim.x + threadIdx.x;
    if (idx < total) out[idx] = bias[idx & (OUT_F - 1)];
}

// ---- Phase 1: support = x @ W via WMMA f32 16x16x4 -------------------------
// One wave handles a 16-node x 16-outf tile; K=128 -> 32 chained WMMAs.
// A (16x4 f32) layout: lane M = lane%16; VGPR v holds K = 2*(lane/16) + v.
// B (4x16 f32) layout: lane N = lane%16; VGPR v holds K = 2*(lane/16) + v.
// C/D (16x16 f32):     lane N = lane%16; VGPR r holds M = r + 8*(lane/16).
__global__ void support_gemm_kernel(const float* __restrict__ x,
                                    const float* __restrict__ w,
                                    float* __restrict__ support,
                                    int n_nodes) {
    const int wave  = (blockIdx.x * blockDim.x + threadIdx.x) >> 5;
    const int lane  = threadIdx.x & 31;
    const int half  = lane >> 4;      // 0 = lanes 0-15, 1 = lanes 16-31
    const int l     = lane & 15;
    const int node0 = wave * 16;
    if (node0 >= n_nodes) return;     // wave-uniform: EXEC all-1s inside

    const float* __restrict__ xrow = x + (size_t)(node0 + l) * IN_F;

    v8f c = {};
#pragma unroll
    for (int k = 0; k < IN_F; k += 4) {
        const int ka = k + 2 * half;
        v2f a, b;
        a.x = xrow[ka + 0];
        a.y = xrow[ka + 1];
        b.x = w[(ka + 0) * OUT_F + l];
        b.y = w[(ka + 1) * OUT_F + l];
        // D = A(16x4) x B(4x16) + C, fp32 throughout
        c = __builtin_amdgcn_wmma_f32_16x16x4_f32(
                /*neg_a=*/false, a, /*neg_b=*/false, b,
                /*c_mod=*/(short)0, c, /*reuse_a=*/false, /*reuse_b=*/false);
    }

    // Store the 16x16 tile: VGPR r -> row (r + 8*half), column l. Each store
    // writes two 64B-contiguous rows across the wave (coalesced).
    float* __restrict__ srow = support + (size_t)node0 * OUT_F;
#pragma unroll
    for (int r = 0; r < 8; ++r)
        srow[(size_t)(r + 8 * half) * OUT_F + l] = c[r];
}

// ---- Phase 2: per-edge matvec + atomic scatter -----------------------------
// 2 edges per wave (lanes 0-15 -> edge0, 16-31 -> edge1); lane = output row i.
// Each lane streams its 64-byte matrix row with 4 non-temporal float4 loads
// (edge_data is 512 MB read once: keep it out of L2 so support/out stay hot),
// broadcasts the gathered source vector within its half-wave via __shfl,
// then atomically accumulates into out[tgt][i].
__global__ void edge_scatter_kernel(const float* __restrict__ support,
                                    const float* __restrict__ edge_data,
                                    const int* __restrict__ Esrc,
                                    const int* __restrict__ Etgt,
                                    float* __restrict__ out,
                                    int n_edges) {
    const int wave = (blockIdx.x * blockDim.x + threadIdx.x) >> 5;
    const int lane = threadIdx.x & 31;
    const int half = lane >> 4;
    const int i    = lane & 15;
    const int e    = wave * 2 + half;
    if (e >= n_edges) return;

    const int src = Esrc[e];
    const int tgt = Etgt[e];

    // Gather one element of the source support vector; rest come via shuffle.
    const float vi = support[(size_t)src * OUT_F + i];

    const vf4* __restrict__ Mrow =
        (const vf4*)(edge_data + (size_t)e * (OUT_F * OUT_F) + (size_t)i * OUT_F);

    float acc = 0.0f;
#pragma unroll
    for (int cidx = 0; cidx < 4; ++cidx) {
        const vf4 m = __builtin_nontemporal_load(&Mrow[cidx]);
        const int base = half * 16 + cidx * 4;
        acc += m.x * __shfl(vi, base + 0, 32);
        acc += m.y * __shfl(vi, base + 1, 32);
        acc += m.z * __shfl(vi, base + 2, 32);
        acc += m.w * __shfl(vi, base + 3, 32);
    }

    atomicAdd(&out[(size_t)tgt * OUT_F + i], acc);
}

// ---------------------------------------------------------------------------
extern "C" void kernel_launch(void* const* d_in, const int* in_sizes, int n_in,
                              void* d_out, int out_size, void* d_ws, size_t ws_size,
                              hipStream_t stream) {
    const float* x         = (const float*)d_in[0];
    const float* weight    = (const float*)d_in[1];
    const float* bias      = (const float*)d_in[2];
    const float* edge_data = (const float*)d_in[3];
    const int*   Esrc      = (const int*)d_in[4];
    const int*   Etgt      = (const int*)d_in[5];
    float*       out       = (float*)d_out;
    float*       support   = (float*)d_ws;   // [n_nodes][16] fp32 scratch

    const int n_nodes = in_sizes[0] / IN_F;  // 50000
    const int n_edges = in_sizes[4];         // 500000

    // Phase 0: out = bias (broadcast); also clears poisoned output.
    {
        const int total  = n_nodes * OUT_F;
        const int blocks = (total + 255) / 256;
        init_bias_kernel<<<blocks, 256, 0, stream>>>(out, bias, total);
    }

    // Phase 1: support = x @ W   (WMMA f32 16x16x4; one wave per 16-node tile)
    {
        const int tiles  = (n_nodes + 15) / 16;      // 3125
        const int blocks = (tiles + 7) / 8;          // 8 waves / 256-thread block
        support_gemm_kernel<<<blocks, 256, 0, stream>>>(x, weight, support, n_nodes);
    }

    // Phase 2: per-edge matvec + atomic segment-sum (bandwidth-bound stream).
    {
        const int pairs  = (n_edges + 1) / 2;        // 250000 waves
        const int blocks = (pairs + 7) / 8;          // 8 waves / block
        edge_scatter_kernel<<<blocks, 256, 0, stream>>>(support, edge_data,
                                                        Esrc, Etgt, out, n_edges);
    }
}